// SparseConvolution_48584670053120
// MI455X (gfx1250) — compile-verified
//
#include <hip/hip_runtime.h>

typedef __attribute__((ext_vector_type(2))) float v2f;
typedef __attribute__((ext_vector_type(8))) float v8f;

#define C_IN  64
#define C_OUT 64
#define LDS_STRIDE 68   // pad so B-frag ds_load_b64 reads are bank-conflict-free

// ---------------------------------------------------------------------------
// out[n][c] = bias[c]  (vectorized float4; C_OUT divisible by 4)
// ---------------------------------------------------------------------------
__global__ void bias_init_kernel(float* __restrict__ out,
                                 const float* __restrict__ bias,
                                 int total4) {
  int i = blockIdx.x * blockDim.x + threadIdx.x;
  if (i < total4) {
    int col = (i * 4) & (C_OUT - 1);
    float4 bv = *(const float4*)(bias + col);
    *(float4*)(out + i * 4) = bv;
  }
}

// ---------------------------------------------------------------------------
// One block: offset k = blockIdx.y, 128 pairs. 8 waves, each wave computes a
// 16x64 tile via V_WMMA_F32_16X16X4_F32 and scatter-adds it.
// ---------------------------------------------------------------------------
__launch_bounds__(256)
__global__ void spconv_wmma_kernel(const float* __restrict__ features,
                                   const float* __restrict__ weight,
                                   const int*   __restrict__ pairs,     // [K,2,P]
                                   const int*   __restrict__ pair_num,  // [K]
                                   float*       __restrict__ out,
                                   int P) {
  __shared__ float wlds[C_OUT * LDS_STRIDE];   // transposed weight[k]: wlds[o*68 + c]

  const int k   = blockIdx.y;
  const int tid = threadIdx.x;

  // Cooperative transposed load of weight[k] (16 KB) into LDS.
  // Global reads are fully coalesced (consecutive tid -> consecutive o).
  const float* Wk = weight + (size_t)k * C_IN * C_OUT;
  #pragma unroll
  for (int i = 0; i < (C_IN * C_OUT) / 256; ++i) {
    int idx = i * 256 + tid;          // idx = c*64 + o
    int c = idx >> 6;
    int o = idx & 63;
    wlds[o * LDS_STRIDE + c] = Wk[idx];
  }
  __syncthreads();

  const int Pk   = pair_num[k];
  const int wave = tid >> 5;
  const int lane = tid & 31;
  const int p0   = blockIdx.x * 128 + wave * 16;   // first pair of this wave's tile
  if (p0 >= Pk) return;                            // wave-uniform exit (after barrier)

  const int row   = lane & 15;        // M row within tile (A layout)
  const int lhalf = lane >> 4;        // 0: K={0,1}, 1: K={2,3} (A layout)
  const int koff  = lhalf * 2;

  const int* pin  = pairs + (size_t)k * 2 * P;   // in indices
  const int* pout = pin + P;                     // out indices

  // ---- A-side gather setup (arithmetic masking keeps EXEC all-ones) ----
  int   p    = p0 + row;
  bool  pv   = p < Pk;
  int   pc   = pv ? p : 0;
  int   ini  = pin[pc];
  float msk  = pv ? 1.0f : 0.0f;
  const float* arow = features + (size_t)ini * C_IN + koff;

  v8f acc0 = {}, acc1 = {}, acc2 = {}, acc3 = {};

  const float* wbase = wlds + row * LDS_STRIDE + koff;

  // 64 iterations of K: 16 steps of K=4, across 4 column tiles of 16.
  #pragma unroll
  for (int s = 0; s < 16; ++s) {
    v2f a = *(const v2f*)(arow + 4 * s);
    a *= msk;                                       // zero rows past Pk
    const float* wb = wbase + 4 * s;
    v2f b0 = *(const v2f*)(wb +  0 * 16 * LDS_STRIDE);
    v2f b1 = *(const v2f*)(wb +  1 * 16 * LDS_STRIDE);
    v2f b2 = *(const v2f*)(wb +  2 * 16 * LDS_STRIDE);
    v2f b3 = *(const v2f*)(wb +  3 * 16 * LDS_STRIDE);
    acc0 = __builtin_amdgcn_wmma_f32_16x16x4_f32(false, a, false, b0, (short)0, acc0, false, false);
    acc1 = __builtin_amdgcn_wmma_f32_16x16x4_f32(false, a, false, b1, (short)0, acc1, false, false);
    acc2 = __builtin_amdgcn_wmma_f32_16x16x4_f32(false, a, false, b2, (short)0, acc2, false, false);
    acc3 = __builtin_amdgcn_wmma_f32_16x16x4_f32(false, a, false, b3, (short)0, acc3, false, false);
  }

  // ---- Scatter-add: C/D layout -> VGPR v holds row (v + 8*lhalf), col = lane&15
  const int col = lane & 15;
  #pragma unroll
  for (int v = 0; v < 8; ++v) {
    int  pm  = p0 + lhalf * 8 + v;
    bool ok  = pm < Pk;
    int  pmc = ok ? pm : 0;
    int  oi  = pout[pmc];
    float* orow = out + (size_t)oi * C_OUT + col;
    if (ok) {
      unsafeAtomicAdd(orow +  0, acc0[v]);
      unsafeAtomicAdd(orow + 16, acc1[v]);
      unsafeAtomicAdd(orow + 32, acc2[v]);
      unsafeAtomicAdd(orow + 48, acc3[v]);
    }
  }
}

// ---------------------------------------------------------------------------
extern "C" void kernel_launch(void* const* d_in, const int* in_sizes, int n_in,
                              void* d_out, int out_size, void* d_ws, size_t ws_size,
                              hipStream_t stream) {
  const float* features = (const float*)d_in[0];
  const float* weight   = (const float*)d_in[1];
  const float* bias     = (const float*)d_in[2];
  const int*   pairs    = (const int*)d_in[3];
  const int*   pair_num = (const int*)d_in[4];
  // d_in[5] = n_out scalar (device); host derives it from out_size instead.
  float* out = (float*)d_out;

  const int K = in_sizes[4];
  const int P = in_sizes[3] / (2 * K);

  int total4 = out_size / 4;
  bias_init_kernel<<<(total4 + 255) / 256, 256, 0, stream>>>(out, bias, total4);

  dim3 grid((P + 127) / 128, K);
  spconv_wmma_kernel<<<grid, 256, 0, stream>>>(features, weight, pairs, pair_num,
                                               out, P);
}